// SubsampledRelativeAttention_197568495697
// MI455X (gfx1250) — compile-verified
//
#include <hip/hip_runtime.h>

typedef float v2f __attribute__((ext_vector_type(2)));
typedef float v8f __attribute__((ext_vector_type(8)));

#define B_  2
#define H_  8
#define L_  2048
#define D_  64
#define NB_ 32            // 2048 / 64 blocks per dim
#define NT_ 528           // NB_*(NB_+1)/2 lower-tri 64x64 blocks

// ---------------------------------------------------------------------------
// Zero-fill the strictly-upper triangle: out[bh, i, k] = 0 for k > i.
// Vectorized float4 stores except at the diagonal boundary.
// ---------------------------------------------------------------------------
__global__ __launch_bounds__(256) void zero_upper_kernel(float* __restrict__ out)
{
    const int row = blockIdx.y;             // 0 .. B_*H_*L_-1  (flat bh*L + i)
    const int i   = row & (L_ - 1);
    const int k4  = (blockIdx.x * 256 + threadIdx.x) * 4;
    if (k4 >= L_) return;
    float* p = out + (size_t)row * L_ + k4;
    if (k4 > i) {
        *(float4*)p = make_float4(0.f, 0.f, 0.f, 0.f);
    } else if (k4 + 3 > i) {
        #pragma unroll
        for (int s = 0; s < 4; ++s)
            if (k4 + s > i) p[s] = 0.0f;
    }
}

// ---------------------------------------------------------------------------
// One wave computes a 64x64 lower-triangular block of G = Q @ E_rev^T
// (K = 64, fp32 WMMA 16x16x4) and scatter-stores with the skew:
//   out[i, i-j] = G[i, j]   (only j <= i valid)
// ---------------------------------------------------------------------------
__global__ __launch_bounds__(128) void relskew_gemm_kernel(
    const float* __restrict__ q,   // (B_, H_, L_, D_)
    const float* __restrict__ e,   // (H_, L_, D_)
    float* __restrict__ out)       // (B_, H_, L_, L_)
{
    const int wave = threadIdx.x >> 5;
    const int lane = threadIdx.x & 31;
    const int lrow = lane & 15;
    const int hsel = lane >> 4;

    const int w  = blockIdx.x * 4 + wave;
    const int bh = w / NT_;
    const int t  = w - bh * NT_;
    if (bh >= B_ * H_) return;
    const int hh = bh & (H_ - 1);

    // decode lower-triangular block index t -> (I, J), J <= I
    int I = (int)((sqrtf(8.0f * (float)t + 1.0f) - 1.0f) * 0.5f);
    while ((I + 1) * (I + 2) / 2 <= t) ++I;
    while (I * (I + 1) / 2 > t) --I;
    const int J  = t - I * (I + 1) / 2;
    const int i0 = I * 64;
    const int j0 = J * 64;

    // A fragment bases: 16x4 f32 tile; lane<16 -> K=k,k+1 ; lane>=16 -> K=k+2,k+3
    const float* qbase = q + (size_t)bh * L_ * D_;
    const float* abase[4];
    #pragma unroll
    for (int ti = 0; ti < 4; ++ti)
        abase[ti] = qbase + (size_t)(i0 + ti * 16 + lrow) * D_ + hsel * 2;

    // B fragment bases: B[k][n] = e[L-1-(j0+tile+n), k]; lane n = lane&15,
    // lanes>=16 carry K=k+2,k+3 (mirrors A layout)
    const float* ebase = e + (size_t)hh * L_ * D_;
    const float* bbase[4];
    #pragma unroll
    for (int tj = 0; tj < 4; ++tj)
        bbase[tj] = ebase + (size_t)(L_ - 1 - (j0 + tj * 16 + lrow)) * D_ + hsel * 2;

    v8f c[4][4];
    #pragma unroll
    for (int ti = 0; ti < 4; ++ti)
        #pragma unroll
        for (int tj = 0; tj < 4; ++tj)
            c[ti][tj] = {};

    #pragma unroll
    for (int kc = 0; kc < D_; kc += 4) {
        v2f a[4], b[4];
        #pragma unroll
        for (int ti = 0; ti < 4; ++ti)
            a[ti] = *(const v2f*)(abase[ti] + kc);
        #pragma unroll
        for (int tj = 0; tj < 4; ++tj)
            b[tj] = *(const v2f*)(bbase[tj] + kc);
        #pragma unroll
        for (int ti = 0; ti < 4; ++ti)
            #pragma unroll
            for (int tj = 0; tj < 4; ++tj)
                c[ti][tj] = __builtin_amdgcn_wmma_f32_16x16x4_f32(
                    false, a[ti], false, b[tj], (short)0, c[ti][tj],
                    false, false);
    }

    // Skew scatter-store: C layout = lane n (=lane&15) holds N=n; VGPR v holds
    // M = v + 8*hsel. out[i, k=i-j], guard k >= 0 for diagonal blocks.
    float* obase = out + (size_t)bh * L_ * L_;
    #pragma unroll
    for (int ti = 0; ti < 4; ++ti) {
        const int ib = i0 + ti * 16 + hsel * 8;
        #pragma unroll
        for (int tj = 0; tj < 4; ++tj) {
            const int jb = j0 + tj * 16 + lrow;
            #pragma unroll
            for (int v = 0; v < 8; ++v) {
                const int i = ib + v;
                const int k = i - jb;
                if (k >= 0)
                    obase[(size_t)i * L_ + k] = c[ti][tj][v];
            }
        }
    }
}

extern "C" void kernel_launch(void* const* d_in, const int* in_sizes, int n_in,
                              void* d_out, int out_size, void* d_ws, size_t ws_size,
                              hipStream_t stream) {
    (void)in_sizes; (void)n_in; (void)out_size; (void)d_ws; (void)ws_size;
    const float* q = (const float*)d_in[0];
    const float* e = (const float*)d_in[1];
    float* out = (float*)d_out;

    // 1) zero the strictly-upper triangle (harness poisons d_out)
    dim3 zgrid(L_ / (4 * 256), B_ * H_ * L_);
    zero_upper_kernel<<<zgrid, 256, 0, stream>>>(out);

    // 2) WMMA GEMM + skew scatter over lower-triangular blocks
    const int total_waves = B_ * H_ * NT_;     // 8448
    const int blocks = (total_waves + 3) / 4;  // 4 waves (128 thr) per block
    relskew_gemm_kernel<<<blocks, 128, 0, stream>>>(q, e, out);
}